// GraffScheduler_71322226917401
// MI455X (gfx1250) — compile-verified
//
#include <hip/hip_runtime.h>
#include <hip/hip_bf16.h>
#include <math.h>

// GraffScheduler fused implementation for gfx1250 (MI455X).
// Key rewrite: all-pairs message passing collapses to
//   agg[j] = S @ W1^T + N * (h'[j] @ W2^T) + N*b_msg,  S = sum_i h'[i]
// turning a 16-GFLOP edge GEMM into an 8-MFLOP node GEMM. All three GEMMs
// (agg, gi, gh) run on V_WMMA_F32_16X16X4_F32 with LDS-staged tiles.

typedef float v2f __attribute__((ext_vector_type(2)));
typedef float v8f __attribute__((ext_vector_type(8)));

#define NNODES 1024
#define FEAT   64
#define MROWS  32    // rows of h per block in the main kernel
#define HS     68    // LDS stride for h'/agg tiles (avoids 64-bank conflicts)
#define GS     260   // LDS stride for gate buffer [r_pre | z_pre | gi_n | gh_n]

static __device__ __forceinline__ v8f wmma_f32_16x16x4(v2f a, v2f b, v8f c) {
  // 8 args: (neg_a, A, neg_b, B, c_mod, C, reuse_a, reuse_b)
  return __builtin_amdgcn_wmma_f32_16x16x4_f32(false, a, false, b, (short)0, c,
                                               false, false);
}

static __device__ __forceinline__ float sigmoidf_(float x) {
  return 1.0f / (1.0f + expf(-x));
}

// ---------------------------------------------------------------------------
// K0: zero the two 64-float accumulators in workspace.
__global__ void k_zero(float* __restrict__ S, float* __restrict__ colsum) {
  int t = threadIdx.x;
  if (t < 64)       S[t] = 0.0f;
  else if (t < 128) colsum[t - 64] = 0.0f;
}

// ---------------------------------------------------------------------------
// K1: h' = h + feat; column sums S += sum over rows of h'.
// 64 blocks x 256 threads; each block owns 16 rows.
__global__ __launch_bounds__(256) void k_addfeat_colsum(
    const float* __restrict__ h, const float* __restrict__ feat,
    float* __restrict__ hprime, float* __restrict__ S) {
  __shared__ float red[256];
  int tid = threadIdx.x;
  int col = tid & 63;
  int r0  = tid >> 6;                 // 0..3
  int rowBase = blockIdx.x * 16;
  float f = feat[col];
  float acc = 0.0f;
  for (int rr = 0; rr < 4; ++rr) {
    int row = rowBase + r0 * 4 + rr;
    float v = h[row * FEAT + col] + f;
    hprime[row * FEAT + col] = v;
    acc += v;
  }
  red[tid] = acc;
  __syncthreads();
  if (r0 == 0) {
    float t = red[col] + red[col + 64] + red[col + 128] + red[col + 192];
    atomicAdd(&S[col], t);
  }
}

// ---------------------------------------------------------------------------
// K2 (main): per 32-row tile, three WMMA f32 GEMMs + fused GRU elementwise.
// 32 blocks x 256 threads (8 waves; wave32).
__global__ __launch_bounds__(256) void k_main(
    const float* __restrict__ hprime, const float* __restrict__ S,
    const float* __restrict__ W_msg, const float* __restrict__ b_msg,
    const float* __restrict__ W_ih,  const float* __restrict__ W_hh,
    const float* __restrict__ b_ih,  const float* __restrict__ b_hh,
    float* __restrict__ h_new, float* __restrict__ colsum) {
  __shared__ float hT[MROWS * HS];      // h' tile           (8.5 KB)
  __shared__ float aggT[MROWS * HS];    // agg tile          (8.5 KB)
  __shared__ float G[MROWS * GS];       // gate buffer       (32.5 KB)
  __shared__ float cC[64];              // c = S@W1^T + N*b_msg
  __shared__ float red[256];

  int tid  = threadIdx.x;
  int lane = tid & 31;
  int w    = tid >> 5;                  // wave id 0..7
  int half = lane >> 4;                 // 0/1: K-pair / M-half selector
  int l16  = lane & 15;
  int rowBase = blockIdx.x * MROWS;

  // Warm L2/WGP$ for the weight matrices (global_prefetch_b8).
  __builtin_prefetch(W_ih + tid, 0, 1);
  __builtin_prefetch(W_hh + tid, 0, 1);

  // Stage h' tile into LDS (row-major, stride HS).
  for (int i = tid; i < MROWS * FEAT; i += 256) {
    int m = i >> 6, c = i & 63;
    hT[m * HS + c] = hprime[(rowBase + m) * FEAT + c];
  }
  // c[n] = S . W_msg[n, 0:64] + 1024 * b_msg[n]
  if (tid < 64) {
    float acc = 0.0f;
    for (int k = 0; k < 64; ++k) acc += S[k] * W_msg[tid * 128 + k];
    cC[tid] = acc + 1024.0f * b_msg[tid];
  }
  __syncthreads();

  // ---- GEMM1: agg = 1024 * (h' @ W2^T) + c. 8 tiles, one per wave. -------
  {
    int mt = w >> 2, nt = w & 3;
    int mBase = mt * 16, nBase = nt * 16;
    v8f acc = {};
    for (int kb = 0; kb < 16; ++kb) {
      int k0 = kb * 4 + half * 2;
      v2f a, b;
      a.x = hT[(mBase + l16) * HS + k0];
      a.y = hT[(mBase + l16) * HS + k0 + 1];
      // B[k][n] = W2[n][k] = W_msg[n*128 + 64 + k]
      const float* wp = W_msg + (nBase + l16) * 128 + 64 + k0;
      b.x = wp[0];
      b.y = wp[1];
      acc = wmma_f32_16x16x4(a, b, acc);
    }
    for (int v = 0; v < 8; ++v) {
      int m = mBase + v + 8 * half;
      aggT[m * HS + nBase + l16] = acc[v] * 1024.0f + cC[nBase + l16];
    }
  }
  __syncthreads();

  // ---- GEMM2: gi = agg @ W_ih^T -> G[:, 0:192]. 24 tiles, 3 per wave. ----
  for (int s = 0; s < 3; ++s) {
    int tile = w + 8 * s;               // 0..23
    int mt = tile / 12, nt = tile % 12;
    int mBase = mt * 16, nBase = nt * 16;
    v8f acc = {};
    for (int kb = 0; kb < 16; ++kb) {
      int k0 = kb * 4 + half * 2;
      v2f a, b;
      a.x = aggT[(mBase + l16) * HS + k0];
      a.y = aggT[(mBase + l16) * HS + k0 + 1];
      const float* wp = W_ih + (nBase + l16) * 64 + k0;   // B[k][n]=W_ih[n][k]
      b.x = wp[0];
      b.y = wp[1];
      acc = wmma_f32_16x16x4(a, b, acc);
    }
    for (int v = 0; v < 8; ++v) {
      int m = mBase + v + 8 * half;
      G[m * GS + nBase + l16] = acc[v];
    }
  }
  __syncthreads();

  // ---- GEMM3: gh = h' @ W_hh^T. r/z gates accumulate, n gate -> +64. -----
  for (int s = 0; s < 3; ++s) {
    int tile = w + 8 * s;
    int mt = tile / 12, nt = tile % 12;
    int mBase = mt * 16, nBase = nt * 16;
    v8f acc = {};
    for (int kb = 0; kb < 16; ++kb) {
      int k0 = kb * 4 + half * 2;
      v2f a, b;
      a.x = hT[(mBase + l16) * HS + k0];
      a.y = hT[(mBase + l16) * HS + k0 + 1];
      const float* wp = W_hh + (nBase + l16) * 64 + k0;
      b.x = wp[0];
      b.y = wp[1];
      acc = wmma_f32_16x16x4(a, b, acc);
    }
    if (nt < 8) {                       // r/z pre-activations: gi + gh
      for (int v = 0; v < 8; ++v) {
        int m = mBase + v + 8 * half;
        G[m * GS + nBase + l16] += acc[v];
      }
    } else {                            // gh_n stored separately at cols 192+
      for (int v = 0; v < 8; ++v) {
        int m = mBase + v + 8 * half;
        G[m * GS + 64 + nBase + l16] = acc[v];
      }
    }
  }
  __syncthreads();

  // ---- Fused GRU elementwise + h_new column partial sums. ----------------
  int col = tid & 63;
  int ml  = tid >> 6;                   // 0..3
  float bR  = b_ih[col]       + b_hh[col];
  float bZ  = b_ih[64 + col]  + b_hh[64 + col];
  float biN = b_ih[128 + col];
  float bhN = b_hh[128 + col];
  float csum = 0.0f;
  for (int rr = 0; rr < 8; ++rr) {
    int m = ml + rr * 4;                // 0..31
    float r  = sigmoidf_(G[m * GS + col] + bR);
    float z  = sigmoidf_(G[m * GS + 64 + col] + bZ);
    float nn = tanhf(G[m * GS + 128 + col] + biN +
                     r * (G[m * GS + 192 + col] + bhN));
    float hv = hT[m * HS + col];
    float out = (1.0f - z) * nn + z * hv;
    h_new[(rowBase + m) * FEAT + col] = out;
    csum += out;
  }
  red[tid] = csum;
  __syncthreads();
  if (ml == 0) {
    float t = red[col] + red[col + 64] + red[col + 128] + red[col + 192];
    atomicAdd(&colsum[col], t);
  }
}

// ---------------------------------------------------------------------------
// K3: lr = clip(exp(mean(h_new,0) @ W_dec^T + b_dec), 1e-3, 10)
__global__ void k_lr(const float* __restrict__ colsum,
                     const float* __restrict__ W_dec,
                     const float* __restrict__ b_dec,
                     float* __restrict__ out_lr) {
  __shared__ float red[64];
  int t = threadIdx.x;
  red[t] = (colsum[t] * (1.0f / 1024.0f)) * W_dec[t];
  __syncthreads();
  if (t == 0) {
    float s = 0.0f;
    for (int i = 0; i < 64; ++i) s += red[i];
    float lr = expf(s + b_dec[0]);
    lr = fminf(fmaxf(lr, 1e-3f), 10.0f);
    out_lr[0] = lr;
  }
}

// ---------------------------------------------------------------------------
extern "C" void kernel_launch(void* const* d_in, const int* in_sizes, int n_in,
                              void* d_out, int out_size, void* d_ws,
                              size_t ws_size, hipStream_t stream) {
  const float* h     = (const float*)d_in[0];
  const float* feat  = (const float*)d_in[1];
  const float* W_msg = (const float*)d_in[2];
  const float* b_msg = (const float*)d_in[3];
  const float* W_ih  = (const float*)d_in[4];
  const float* W_hh  = (const float*)d_in[5];
  const float* b_ih  = (const float*)d_in[6];
  const float* b_hh  = (const float*)d_in[7];
  const float* W_dec = (const float*)d_in[8];
  const float* b_dec = (const float*)d_in[9];

  float* out = (float*)d_out;           // 65536 floats h_new, then 1 float lr
  float* ws  = (float*)d_ws;
  float* hprime = ws;                   // 65536 floats
  float* S      = ws + 65536;           // 64 floats
  float* colsum = ws + 65600;           // 64 floats

  k_zero<<<1, 128, 0, stream>>>(S, colsum);
  k_addfeat_colsum<<<64, 256, 0, stream>>>(h, feat, hprime, S);
  k_main<<<32, 256, 0, stream>>>(hprime, S, W_msg, b_msg, W_ih, W_hh,
                                 b_ih, b_hh, out, colsum);
  k_lr<<<1, 64, 0, stream>>>(colsum, W_dec, b_dec, out + 65536);
}